// Spatial_tansform_tu_6743098655507
// MI455X (gfx1250) — compile-verified
//
#include <hip/hip_runtime.h>
#include <hip/hip_bf16.h>

typedef float v2f __attribute__((ext_vector_type(2)));
typedef float v8f __attribute__((ext_vector_type(8)));
typedef unsigned int v4u __attribute__((ext_vector_type(4)));
typedef int v4i __attribute__((ext_vector_type(4)));
typedef int v8i __attribute__((ext_vector_type(8)));

#define VOL      2097152      // 128^3
#define LL       128
#define RAD      52
#define KLEN     105
#define NCHUNK   30           // ceil((105+15)/4)
#define IN_STR   233          // m-major layout row stride (odd -> conflict-free)
#define OUT_STR  129

#define LSB_FLOATS  (NCHUNK * 32 * 2)          // 1920 floats of band matrices
#define INBUF       3728                       // per-wave input staging (floats)
#define OUTBUF      (16 * OUT_STR)             // per-wave output staging (floats)
#define SMEM_FLOATS (LSB_FLOATS + 2 * INBUF + 2 * OUTBUF)   // 13504 -> 54 KB

#if defined(__has_builtin)
#  if __has_builtin(__builtin_amdgcn_tensor_load_to_lds) && \
      __has_builtin(__builtin_amdgcn_s_wait_tensorcnt)
#    define CONV_USE_TDM 1
#  endif
#endif
#ifndef CONV_USE_TDM
#  define CONV_USE_TDM 0
#endif

// ---------------------------------------------------------------------------
// prep: Gaussian weights, WMMA band matrices, zero-pad correction S, rotation
// ---------------------------------------------------------------------------
__global__ __launch_bounds__(128) void prep_kernel(
    const float* __restrict__ alpha_u, const float* __restrict__ sigma_u,
    const float* __restrict__ angle_u, const float* __restrict__ scale_u,
    float* __restrict__ params, float* __restrict__ Sarr, v2f* __restrict__ Bl)
{
    int b = blockIdx.x;
    int tid = threadIdx.x;
    __shared__ float red[128];
    __shared__ float kloc[112];

    float sigma = 10.f + 3.f * sigma_u[b];
    float e = 0.f;
    if (tid < KLEN) {
        float t = (float)tid - 52.f;
        float q = t / sigma;
        e = expf(-0.5f * q * q);
    }
    red[tid] = e;
    __syncthreads();
    for (int s = 64; s > 0; s >>= 1) {
        if (tid < s) red[tid] += red[tid + s];
        __syncthreads();
    }
    float inv = 1.f / red[0];
    if (tid < 112) kloc[tid] = (tid < KLEN) ? e * inv : 0.f;
    __syncthreads();

    // 1D zero-padding response: S[p] = sum_t k[t] * [0 <= p+t-52 < 128]
    {
        float s = 0.f;
        for (int t = 0; t < KLEN; ++t) {
            int p = tid + t - RAD;
            if (p >= 0 && p < LL) s += kloc[t];
        }
        Sarr[b * LL + tid] = s;
    }

    // Band matrices in WMMA B lane layout: lane l -> n = l&15, k = (l>>4)*2 (+1)
    int lane = tid & 31;
    int n = lane & 15;
    int kb = (lane >> 4) * 2;
    for (int rep = 0; rep < 8; ++rep) {
        int c = (tid >> 5) + rep * 4;
        if (c < NCHUNK) {
            v2f v;
            int t0 = 4 * c + kb + 0 - n;
            int t1 = 4 * c + kb + 1 - n;
            v.x = (t0 >= 0 && t0 < KLEN) ? kloc[t0] : 0.f;
            v.y = (t1 >= 0 && t1 < KLEN) ? kloc[t1] : 0.f;
            Bl[(size_t)b * (NCHUNK * 32) + c * 32 + lane] = v;
        }
    }

    if (tid == 0) {
        float alpha = alpha_u[b] * 1000.f;
        float sc = 0.75f + 0.25f * scale_u[b];
        const float TWO_PI = 6.283185307179586f;
        float ax = angle_u[b * 3 + 0] * TWO_PI;
        float ay = angle_u[b * 3 + 1] * TWO_PI;
        float az = angle_u[b * 3 + 2] * TWO_PI;
        float cx = cosf(ax), sx = sinf(ax);
        float cy = cosf(ay), sy = sinf(ay);
        float cz = cosf(az), sz = sinf(az);
        float Rx[9] = {1, 0, 0, 0, cx, -sx, 0, sx, cx};
        float Ry[9] = {cy, 0, sy, 0, 1, 0, -sy, 0, cy};
        float Rz[9] = {cz, -sz, 0, sz, cz, 0, 0, 0, 1};
        float T[9], R[9];
        for (int i = 0; i < 3; ++i)
            for (int j = 0; j < 3; ++j) {
                float s = 0.f;
                for (int k = 0; k < 3; ++k) s += Ry[i * 3 + k] * Rz[k * 3 + j];
                T[i * 3 + j] = s;
            }
        for (int i = 0; i < 3; ++i)
            for (int j = 0; j < 3; ++j) {
                float s = 0.f;
                for (int k = 0; k < 3; ++k) s += Rx[i * 3 + k] * T[k * 3 + j];
                R[i * 3 + j] = s;
            }
        float* P = params + b * 16;
        P[0] = alpha;
        P[1] = sc;
        for (int i = 0; i < 9; ++i) P[2 + i] = R[i];  // R[b_axis][a_axis]
    }
}

// ---------------------------------------------------------------------------
// conv_pass: 105-tap 1D convolution along one axis via V_WMMA_F32_16X16X4_F32.
// Tile load from global into LDS via the Tensor Data Mover (TENSOR_LOAD_TO_LDS,
// TENSORcnt), with pad_enable providing the stride-233 LDS layout on the H axis.
// ---------------------------------------------------------------------------
__global__ __launch_bounds__(64) void conv_pass_kernel(
    const float* __restrict__ src, float* __restrict__ dst,
    const v2f* __restrict__ BlG,
    int stride_n, int strideHi, int strideLo, int laneAlongN, int stride0)
{
    __shared__ __align__(16) float smem[SMEM_FLOATS];
    v2f* lsB = (v2f*)smem;

    int tid  = threadIdx.x;
    int wv   = tid >> 5;
    int lane = tid & 31;

    int vol = blockIdx.x >> 9;                  // 512 line-groups per volume
    int grp = blockIdx.x & 511;
    int batch = vol / 3;
    int l0 = grp * 32 + wv * 16;
    int lineBase = (l0 >> 7) * strideHi + (l0 & 127) * strideLo;
    long volBase = (long)vol * VOL;
    int inBase  = LSB_FLOATS + wv * INBUF;
    int outBase = LSB_FLOATS + 2 * INBUF + wv * OUTBUF;
    int layN = !laneAlongN;                     // tap-major LDS layout (W/D axes)

    const float* gsrc = src + volBase + lineBase;

#if CONV_USE_TDM
    {
        // D# group 0: count=1, lds byte addr, 57-bit global addr, type=2
        unsigned long long ga = (unsigned long long)(uintptr_t)gsrc;
        unsigned lds_byte = (unsigned)((inBase + (layN ? RAD * 16 : RAD)) * 4);
        unsigned td0, td1, tl0, tl1, padcfg;
        if (layN) { td0 = 16;  td1 = 128; tl0 = 16;  tl1 = 128; padcfg = 0; }
        else      { td0 = 128; td1 = 16;  tl0 = 128; tl1 = 16;
                    // pad 105 dwords after every 128 dwords -> LDS row stride 233
                    padcfg = (1u << 20) | (6u << 22) | (104u << 25); }
        v4u g0 = { 1u, lds_byte, (unsigned)(ga & 0xFFFFFFFFu),
                   (unsigned)((ga >> 32) & 0x1FFFFFFu) | (2u << 30) };
        v8i g1;
        g1[0] = (int)((2u << 16) | padcfg);                 // data_size=4B
        g1[1] = (int)((td0 & 0xFFFFu) << 16);               // tensor_dim0 lo
        g1[2] = (int)(((td0 >> 16) & 0xFFFFu) | ((td1 & 0xFFFFu) << 16));
        g1[3] = (int)(((td1 >> 16) & 0xFFFFu) | (tl0 << 16));  // tile_dim0
        g1[4] = (int)(tl1 & 0xFFFFu);                       // tile_dim1, dim2=0
        g1[5] = (int)stride0;                               // tensor_dim0_stride
        g1[6] = 0;
        g1[7] = 0;
        v4i g2 = {0, 0, 0, 0};
        v4i g3 = {0, 0, 0, 0};
#if __clang_major__ >= 23
        v8i gx = {0, 0, 0, 0, 0, 0, 0, 0};
        __builtin_amdgcn_tensor_load_to_lds(g0, g1, g2, g3, gx, 0);
#else
        __builtin_amdgcn_tensor_load_to_lds(g0, g1, g2, g3, 0);
#endif
    }
#endif

    // band matrices: cooperative load (same batch for whole block)
    for (int i = tid; i < NCHUNK * 32; i += 64)
        lsB[i] = BlG[(size_t)batch * (NCHUNK * 32) + i];

    // zero-fill the halo regions (disjoint from the TDM destination)
    if (layN) {
        for (int i = lane; i < 2 * 832; i += 32)
            smem[inBase + (i < 832 ? i : i + 2048)] = 0.f;
    } else {
        for (int i = lane; i < 16 * KLEN; i += 32) {
            int m = i / KLEN;
            int c = i % KLEN;
            smem[inBase + m * IN_STR + (c < RAD ? c : c + LL)] = 0.f;
        }
    }

#if CONV_USE_TDM
    __builtin_amdgcn_s_wait_tensorcnt(0);
#else
    for (int i = lane; i < 16 * LL; i += 32) {
        int m, n;
        if (laneAlongN) { m = i >> 7; n = i & 127; }
        else            { m = i & 15; n = i >> 4; }
        float v = gsrc[(long)m * strideLo + (long)n * stride_n];
        smem[inBase + (layN ? (RAD + n) * 16 + m : m * IN_STR + RAD + n)] = v;
    }
#endif
    __syncthreads();

    int mrow = lane & 15;
    int hi   = lane >> 4;
    int kb   = hi * 2;

    if (layN) {
        #pragma unroll 1
        for (int t = 0; t < 8; ++t) {
            int n0 = t * 16;
            v8f acc = {};
            #pragma unroll
            for (int c = 0; c < NCHUNK; ++c) {
                int ai = inBase + (n0 + 4 * c + kb) * 16 + mrow;
                v2f a;
                a.x = smem[ai];
                a.y = smem[ai + 16];
                v2f bb = lsB[c * 32 + lane];
                acc = __builtin_amdgcn_wmma_f32_16x16x4_f32(
                    false, a, false, bb, (short)0, acc, false, false);
            }
            #pragma unroll
            for (int r = 0; r < 8; ++r)
                smem[outBase + (r + hi * 8) * OUT_STR + n0 + mrow] = acc[r];
        }
    } else {
        #pragma unroll 1
        for (int t = 0; t < 8; ++t) {
            int n0 = t * 16;
            v8f acc = {};
            #pragma unroll
            for (int c = 0; c < NCHUNK; ++c) {
                int ai = inBase + mrow * IN_STR + n0 + 4 * c + kb;
                v2f a;
                a.x = smem[ai];
                a.y = smem[ai + 1];
                v2f bb = lsB[c * 32 + lane];
                acc = __builtin_amdgcn_wmma_f32_16x16x4_f32(
                    false, a, false, bb, (short)0, acc, false, false);
            }
            #pragma unroll
            for (int r = 0; r < 8; ++r)
                smem[outBase + (r + hi * 8) * OUT_STR + n0 + mrow] = acc[r];
        }
    }
    __syncthreads();

    // store with coalesced lane mapping for this axis
    for (int i = lane; i < 16 * LL; i += 32) {
        int m, n;
        if (laneAlongN) { m = i >> 7; n = i & 127; }
        else            { m = i & 15; n = i >> 4; }
        dst[volBase + lineBase + (long)m * strideLo + (long)n * stride_n] =
            smem[outBase + m * OUT_STR + n];
    }
}

// ---------------------------------------------------------------------------
// warp_interp: rotate/scale coordinates + 4x4x4 Catmull-Rom resample.
// Applies offs = 2*blur(u) - S(d)S(w)S(h) (separable zero-pad correction).
// ---------------------------------------------------------------------------
__device__ __forceinline__ void cr_axis(float p, float* w, int* ix) {
    float fl = floorf(p);
    float t = p - fl;
    float t2 = t * t, t3 = t2 * t;
    w[0] = 0.5f * (-t3 + 2.f * t2 - t);
    w[1] = 0.5f * (3.f * t3 - 5.f * t2 + 2.f);
    w[2] = 0.5f * (-3.f * t3 + 4.f * t2 + t);
    w[3] = 0.5f * (t3 - t2);
    int fi = (int)fl;
    #pragma unroll
    for (int o = 0; o < 4; ++o) {
        int v = fi - 1 + o;
        ix[o] = v < 0 ? 0 : (v > 127 ? 127 : v);
    }
}

__global__ __launch_bounds__(256) void warp_interp_kernel(
    const float* __restrict__ x, const float* __restrict__ offs,
    const float* __restrict__ params, const float* __restrict__ Sarr,
    float* __restrict__ out)
{
    int idx = blockIdx.x * 256 + threadIdx.x;
    int b = idx >> 21;
    int v = idx & (VOL - 1);
    int d = v >> 14;
    int w = (v >> 7) & 127;
    int h = v & 127;

    const float* P = params + b * 16;
    float alpha = P[0], sc = P[1];
    const float* R = P + 2;
    const float* Sb = Sarr + b * LL;
    float corr = Sb[d] * Sb[w] * Sb[h];

    const float* o0 = offs + (size_t)b * 3 * VOL + v;
    float c0 = ((float)d - 63.5f) + (2.f * o0[0]        - corr) * alpha;
    float c1 = ((float)w - 63.5f) + (2.f * o0[VOL]      - corr) * alpha;
    float c2 = ((float)h - 63.5f) + (2.f * o0[2 * VOL]  - corr) * alpha;

    // new[a] = sum_b c[b] * R[b][a]
    float p0 = (c0 * R[0] + c1 * R[3] + c2 * R[6]) * sc + 63.5f;
    float p1 = (c0 * R[1] + c1 * R[4] + c2 * R[7]) * sc + 63.5f;
    float p2 = (c0 * R[2] + c1 * R[5] + c2 * R[8]) * sc + 63.5f;

    float wz[4], wy[4], wx[4];
    int iz[4], iy[4], ix[4];
    cr_axis(p0, wz, iz);
    cr_axis(p1, wy, iy);
    cr_axis(p2, wx, ix);

    const float* vol = x + (size_t)b * VOL;
    __builtin_prefetch(vol + ((iz[0] * LL + iy[0]) * LL + ix[0]), 0, 0);

    float acc = 0.f;
    #pragma unroll
    for (int i = 0; i < 4; ++i) {
        float wi = wz[i];
        int bi = iz[i] * LL;
        #pragma unroll
        for (int j = 0; j < 4; ++j) {
            float wij = wi * wy[j];
            const float* row = vol + (bi + iy[j]) * LL;
            float s = wx[0] * row[ix[0]] + wx[1] * row[ix[1]] +
                      wx[2] * row[ix[2]] + wx[3] * row[ix[3]];
            acc += wij * s;
        }
    }
    out[idx] = acc;
}

// ---------------------------------------------------------------------------
extern "C" void kernel_launch(void* const* d_in, const int* in_sizes, int n_in,
                              void* d_out, int out_size, void* d_ws, size_t ws_size,
                              hipStream_t stream) {
    const float* x       = (const float*)d_in[0];  // (2,1,128,128,128)
    const float* noise   = (const float*)d_in[1];  // (2,3,128,128,128)
    const float* alpha_u = (const float*)d_in[2];
    const float* sigma_u = (const float*)d_in[3];
    const float* angle_u = (const float*)d_in[4];
    const float* scale_u = (const float*)d_in[5];
    float* out = (float*)d_out;

    float* wsf    = (float*)d_ws;
    float* params = wsf;                       // 2*16 floats
    float* Sarr   = wsf + 32;                  // 2*128 floats
    v2f*   Bl     = (v2f*)(wsf + 512);         // 2*30*32 v2f = 3840 floats
    float* bufA   = wsf + 8192;                // 6*VOL floats
    float* bufB   = bufA + (size_t)6 * VOL;

    prep_kernel<<<dim3(2), dim3(128), 0, stream>>>(
        alpha_u, sigma_u, angle_u, scale_u, params, Sarr, Bl);

    // axis D: lines over (w,h): base = w*128 + h, element stride 16384
    conv_pass_kernel<<<dim3(3072), dim3(64), 0, stream>>>(
        noise, bufA, Bl, 16384, 128, 1, /*laneAlongN=*/0, /*stride0=*/16384);
    // axis W: lines over (d,h): base = d*16384 + h, element stride 128
    conv_pass_kernel<<<dim3(3072), dim3(64), 0, stream>>>(
        bufA, bufB, Bl, 128, 16384, 1, 0, 128);
    // axis H: lines over (d,w): base = d*16384 + w*128, element stride 1
    conv_pass_kernel<<<dim3(3072), dim3(64), 0, stream>>>(
        bufB, bufA, Bl, 1, 16384, 128, 1, 128);

    warp_interp_kernel<<<dim3(2 * VOL / 256), dim3(256), 0, stream>>>(
        x, bufA, params, Sarr, out);
}